// RetinaNet_60181081751585
// MI455X (gfx1250) — compile-verified
//
#include <hip/hip_runtime.h>
#include <cstdint>

#define RN_C 80
#define RN_G 64
#define TILE 128

typedef float v2f __attribute__((ext_vector_type(2)));
typedef float v8f __attribute__((ext_vector_type(8)));

// ---------------------------------------------------------------------------
// Exact 32-lane sum via V_WMMA_F32_16X16X4_F32 with B == ones.
// A layout (16x4 f32): lanes0-15 vgpr0=A[m,0], vgpr1=A[m,1]; lanes16-31 -> K=2,3.
// With a = (v, 0): D[m,n] = v[m] + v[m+16].  D vgpr j = rows {j, j+8}.
// Sum of the 8 D vgprs per lane + shfl_xor(16) == sum over all 32 lanes.
// Requires EXEC all ones: call from fully-converged full waves only.
// ---------------------------------------------------------------------------
__device__ __forceinline__ float wave_sum32(float v) {
    v2f a; a.x = v;    a.y = 0.0f;
    v2f b; b.x = 1.0f; b.y = 1.0f;
    v8f c = {};
    c = __builtin_amdgcn_wmma_f32_16x16x4_f32(false, a, false, b,
                                              (short)0, c, false, false);
    float s = c[0] + c[1] + c[2] + c[3] + c[4] + c[5] + c[6] + c[7];
    s += __shfl_xor(s, 16, 32);
    return s;
}

__device__ __forceinline__ float iou_box(float ax1, float ay1, float ax2, float ay2,
                                         float areaA, float4 g) {
    float areaG = (g.z - g.x) * (g.w - g.y);
    float ix1 = fmaxf(ax1, g.x), iy1 = fmaxf(ay1, g.y);
    float ix2 = fminf(ax2, g.z), iy2 = fminf(ay2, g.w);
    float iw = fmaxf(ix2 - ix1, 0.0f), ih = fmaxf(iy2 - iy1, 0.0f);
    float inter = iw * ih;
    return inter / (areaA + areaG - inter);
}

__device__ __forceinline__ float smooth_l1(float d) {
    float a = fabsf(d);
    return (a < 0.1f) ? (0.5f * d * d * 10.0f) : (a - 0.05f);
}

// Focal term for one logit, split by target t (t in {0,1}).
// e = exp(-|x|), l = log1p(e), p = sigmoid(x).
__device__ __forceinline__ void focal_parts(float x, float& f0, float& f1) {
    float e   = expf(-fabsf(x));
    float l   = log1pf(e);
    float inv = 1.0f / (1.0f + e);          // lowers to rcp+NR, no second expf
    float p   = (x >= 0.0f) ? inv : e * inv;
    float mx  = fmaxf(x, 0.0f);
    // t = 0: ce = mx + l,      1-pt = p,     alpha_t = 0.75
    f0 = (mx + l) * p * p * 0.75f;
    // t = 1: ce = mx - x + l,  1-pt = 1-p,   alpha_t = 0.25
    float q = 1.0f - p;
    f1 = (mx - x + l) * q * q * 0.25f;
}

// --------------------------- K0: init workspace ----------------------------
__global__ void rn_init(unsigned* maxbits, int n) {
    int i = blockIdx.x * blockDim.x + threadIdx.x;
    if (i < n) maxbits[i] = 0u;
}

// ---------------- K1: per-GT max IoU over all anchors (per batch) ----------
__global__ void __launch_bounds__(256)
rn_max_per_gt(const float4* __restrict__ anchors, const float4* __restrict__ gt,
              unsigned* __restrict__ maxbits, int R) {
    __shared__ unsigned sMax[RN_G];
    __shared__ float4   sG[RN_G];
    const int b = blockIdx.y;
    const int tid = threadIdx.x;
    if (tid < RN_G) { sMax[tid] = 0u; sG[tid] = gt[b * RN_G + tid]; }
    __syncthreads();

    const int r = blockIdx.x * 256 + tid;
    const bool ok = (r < R);
    float4 a = anchors[ok ? r : (R - 1)];
    float areaA = (a.z - a.x) * (a.w - a.y);

    for (int g = 0; g < RN_G; ++g) {
        float v = ok ? iou_box(a.x, a.y, a.z, a.w, areaA, sG[g]) : 0.0f;
        #pragma unroll
        for (int off = 16; off; off >>= 1)
            v = fmaxf(v, __shfl_xor(v, off, 32));
        if ((tid & 31) == 0)
            atomicMax(&sMax[g], __float_as_uint(v));
    }
    __syncthreads();
    if (tid < RN_G) atomicMax(&maxbits[b * RN_G + tid], sMax[tid]);
}

// ---------------- K2: fused matcher + focal + smooth-L1 --------------------
__global__ void __launch_bounds__(TILE)
rn_main(const float* __restrict__ logits, const float4* __restrict__ deltas,
        const float4* __restrict__ anchors, const float4* __restrict__ gt,
        const int* __restrict__ labels, const unsigned* __restrict__ maxbits,
        float* __restrict__ partials, int R) {
    __shared__ float  sLog[TILE * RN_C];   // 40 KB staged tile of pred_logits
    __shared__ float4 sG[RN_G];
    __shared__ int    sLab[RN_G];
    __shared__ float  sMaxPg[RN_G];
    __shared__ float  sPart[TILE / 32][3];

    const int b   = blockIdx.y;
    const int tid = threadIdx.x;
    const int tileBase = blockIdx.x * TILE;
    const int rows = min(TILE, R - tileBase);

    if (tid < RN_G) {
        sG[tid]     = gt[b * RN_G + tid];
        sLab[tid]   = labels[b * RN_G + tid];
        sMaxPg[tid] = __uint_as_float(maxbits[b * RN_G + tid]);
    }

    // --- async stage: contiguous [rows x 80] f32 tile -> LDS (ASYNCcnt path)
    {
        const uint64_t gbase = (uint64_t)(uintptr_t)(logits) +
                               ((uint64_t)b * R + (uint64_t)tileBase) * RN_C * 4u;
        const unsigned ldsBase = (unsigned)(uintptr_t)(void*)sLog; // low 32 bits = LDS offset
        const int nvec = rows * (RN_C / 4);                        // 16B chunks
        for (int i = tid; i < nvec; i += TILE) {
            unsigned off = (unsigned)i * 16u;
            asm volatile("global_load_async_to_lds_b128 %0, %1, %2"
                         :: "v"(ldsBase + off), "v"(off), "s"(gbase)
                         : "memory");
        }
        asm volatile("s_wait_asynccnt 0" ::: "memory");
    }
    __syncthreads();

    const int r = tileBase + tid;
    float cls = 0.0f, box = 0.0f, nfg = 0.0f;

    if (r < R) {
        float4 a = anchors[r];
        float areaA = (a.z - a.x) * (a.w - a.y);

        float best = -1.0f;  // first-argmax tie-break (strict >)
        int   bm   = 0;
        bool  lq   = false;
        for (int g = 0; g < RN_G; ++g) {
            float v = iou_box(a.x, a.y, a.z, a.w, areaA, sG[g]);
            if (v > best) { best = v; bm = g; }
            float mpg = sMaxPg[g];
            lq = lq || ((v == mpg) && (mpg > 0.0f));
        }
        bool fg  = (best >= 0.5f) || lq;
        bool ign = (!fg) && (best >= 0.4f);
        bool val = fg || (!ign);
        int  lbl = fg ? sLab[bm] : -1;

        if (val) {
            const float* row = &sLog[tid * RN_C];
            // hot loop: pure t=0 specialization, no per-class selects
            float s = 0.0f;
            for (int c = 0; c < RN_C; ++c) {
                float f0, f1;
                focal_parts(row[c], f0, f1);
                s += f0;
            }
            // single-class correction for the matched (foreground) label
            if (lbl >= 0) {
                float f0, f1;
                focal_parts(row[lbl], f0, f1);
                s += f1 - f0;
            }
            cls = s;
        }
        if (fg) {
            float4 gb = sG[bm];
            float4 d  = deltas[(size_t)b * R + r];
            float sw = a.z - a.x,  sh = a.w - a.y;
            float scx = a.x + 0.5f * sw, scy = a.y + 0.5f * sh;
            float tw = gb.z - gb.x, th = gb.w - gb.y;
            float tcx = gb.x + 0.5f * tw, tcy = gb.y + 0.5f * th;
            float t0 = (tcx - scx) / sw, t1 = (tcy - scy) / sh;
            float t2 = logf(tw / sw),    t3 = logf(th / sh);
            box = smooth_l1(d.x - t0) + smooth_l1(d.y - t1) +
                  smooth_l1(d.z - t2) + smooth_l1(d.w - t3);
            nfg = 1.0f;
        }
    }

    // fully-converged wave reductions (WMMA f32 16x16x4, EXEC all-ones)
    float wc = wave_sum32(cls);
    float wb = wave_sum32(box);
    float wn = wave_sum32(nfg);
    const int wave = tid >> 5;
    if ((tid & 31) == 0) { sPart[wave][0] = wc; sPart[wave][1] = wb; sPart[wave][2] = wn; }
    __syncthreads();
    if (tid == 0) {
        float c0 = 0.0f, b0 = 0.0f, n0 = 0.0f;
        for (int w = 0; w < TILE / 32; ++w) { c0 += sPart[w][0]; b0 += sPart[w][1]; n0 += sPart[w][2]; }
        const int blk = blockIdx.y * gridDim.x + blockIdx.x;   // fixed slot: deterministic
        partials[blk * 3 + 0] = c0;
        partials[blk * 3 + 1] = b0;
        partials[blk * 3 + 2] = n0;
    }
}

// ---------------- K3: deterministic final reduction ------------------------
__global__ void __launch_bounds__(256)
rn_final(const float* __restrict__ partials, int nblk, float* __restrict__ out) {
    __shared__ float s[256][3];
    const int tid = threadIdx.x;
    float c = 0.0f, bx = 0.0f, n = 0.0f;
    for (int i = tid; i < nblk; i += 256) {
        c  += partials[i * 3 + 0];
        bx += partials[i * 3 + 1];
        n  += partials[i * 3 + 2];
    }
    s[tid][0] = c; s[tid][1] = bx; s[tid][2] = n;
    __syncthreads();
    for (int off = 128; off; off >>= 1) {
        if (tid < off) {
            s[tid][0] += s[tid + off][0];
            s[tid][1] += s[tid + off][1];
            s[tid][2] += s[tid + off][2];
        }
        __syncthreads();
    }
    if (tid == 0) {
        float norm = fmaxf(1.0f, s[0][2]);
        out[0] = s[0][0] / norm;
        out[1] = s[0][1] / norm;
    }
}

// ---------------------------------------------------------------------------
extern "C" void kernel_launch(void* const* d_in, const int* in_sizes, int n_in,
                              void* d_out, int out_size, void* d_ws, size_t ws_size,
                              hipStream_t stream) {
    const float*  logits  = (const float*)d_in[0];   // [B,R,80]
    const float4* deltas  = (const float4*)d_in[1];  // [B,R,4]
    const float4* anchors = (const float4*)d_in[2];  // [R,4]
    const float4* gtb     = (const float4*)d_in[3];  // [B,64,4]
    const int*    gtl     = (const int*)d_in[4];     // [B,64]

    const int R = in_sizes[2] / 4;
    const int B = in_sizes[4] / RN_G;

    unsigned* maxbits = (unsigned*)d_ws;
    float* partials = (float*)((char*)d_ws + (((size_t)B * RN_G * sizeof(unsigned) + 255) & ~(size_t)255));

    const int nMB  = B * RN_G;
    rn_init<<<(nMB + 255) / 256, 256, 0, stream>>>(maxbits, nMB);

    dim3 g1((R + 255) / 256, B);
    rn_max_per_gt<<<g1, 256, 0, stream>>>(anchors, gtb, maxbits, R);

    dim3 g2((R + TILE - 1) / TILE, B);
    rn_main<<<g2, TILE, 0, stream>>>(logits, deltas, anchors, gtb, gtl,
                                     maxbits, partials, R);

    const int nblk = g2.x * B;
    rn_final<<<1, 256, 0, stream>>>(partials, nblk, (float*)d_out);
}